// ForceThresholdCost_69655779606608
// MI455X (gfx1250) — compile-verified
//
#include <hip/hip_runtime.h>
#include <hip/hip_bf16.h>

typedef __attribute__((ext_vector_type(2))) float v2f;
typedef __attribute__((ext_vector_type(8))) float v8f;

#define THRESH  3.0f
#define PENALTY 10000.0f

// ---------------------------------------------------------------------------
// Kernel 1: collapse normals/jac/forces/start_state into w[0..6] and F'.
//   w[k] = 1000 * sum_c (sum_i normals[c,i]) * (sum_j jac[c,j,k])
//   F'   = sum_c forces[c] - dot(w, start_state)
// ws[0..6] = w, ws[7] = F'
// ---------------------------------------------------------------------------
__global__ void prep_kernel(const float* __restrict__ start_state,
                            const float* __restrict__ normals,
                            const float* __restrict__ jac,
                            const float* __restrict__ forces,
                            float* __restrict__ ws, int C) {
    __shared__ float part[64][8];
    __shared__ float wsm[8];
    int c = threadIdx.x;  // 64 threads, one per constraint
    float n = 0.0f, f = 0.0f;
    float pk[7];
    if (c < C) {
        n = normals[c * 3 + 0] + normals[c * 3 + 1] + normals[c * 3 + 2];
        f = forces[c];
    }
#pragma unroll
    for (int k = 0; k < 7; ++k) {
        float j = 0.0f;
        if (c < C)
            j = jac[c * 21 + k] + jac[c * 21 + 7 + k] + jac[c * 21 + 14 + k];
        pk[k] = n * j;
    }
#pragma unroll
    for (int k = 0; k < 7; ++k) part[c][k] = pk[k];
    part[c][7] = f;
    __syncthreads();

    if (threadIdx.x < 8) {
        int k = threadIdx.x;
        float s = 0.0f;
        for (int cc = 0; cc < 64; ++cc) s += part[cc][k];
        wsm[k] = (k < 7) ? (1000.0f * s) : s;
    }
    __syncthreads();

    if (threadIdx.x == 0) {
        float fp = wsm[7];
#pragma unroll
        for (int k = 0; k < 7; ++k) {
            fp -= wsm[k] * start_state[k];
            ws[k] = wsm[k];
        }
        ws[7] = fp;
    }
}

// ---------------------------------------------------------------------------
// Kernel 2: streaming pass. Each wave32 handles tiles of 16 rows.
// A (16x4 f32): 16 rows of state_batch (K slice). Per ISA layout:
//   lanes 0-15 hold M=lane, VGPR0=K0, VGPR1=K1; lanes 16-31: K2, K3.
// B (4x16 f32): every column = w  (lanes 0-15: {wK0,wK1}; 16-31: {wK2,wK3}).
// C seeded with F'.  Two chained WMMAs cover K=0..7 (elem 7 = 0).
// Column 0 of D = fi for the 16 rows; lanes 0/16 threshold + store.
// ---------------------------------------------------------------------------
__global__ __launch_bounds__(256)
void cost_wmma_kernel(const float* __restrict__ sb,
                      const float* __restrict__ ws,
                      float* __restrict__ out, int ntiles) {
    float w[8];
#pragma unroll
    for (int k = 0; k < 7; ++k) w[k] = ws[k];
    w[7] = 0.0f;
    const float Fp = ws[7];

    const int lane  = threadIdx.x & 31;
    const int half  = lane >> 4;       // 0: K0/K1 group, 1: K2/K3 group
    const int row16 = lane & 15;       // M within the tile
    const int wave  = (int)((blockIdx.x * blockDim.x + threadIdx.x) >> 5);
    const int nwav  = (int)((gridDim.x * blockDim.x) >> 5);

    v2f b0, b1;
    b0.x = w[2 * half];     b0.y = w[2 * half + 1];     // K=0..3
    b1.x = w[4 + 2 * half]; b1.y = w[5 + 2 * half];     // K=4..7 (w[7]=0)

    v8f cinit;
#pragma unroll
    for (int i = 0; i < 8; ++i) cinit[i] = Fp;

    for (int tile = wave; tile < ntiles; tile += nwav) {
        const float* rowp = sb + (size_t)(tile * 16 + row16) * 7;
        // All loads unconditional (rowp[0..6] always in-bounds for the row);
        // the K=7 pad is applied with a register select, NOT a masked load,
        // so the inner loop carries no EXEC manipulation before the WMMAs.
        v2f a0, a1;
        a0.x = rowp[2 * half];
        a0.y = rowp[2 * half + 1];
        a1.x = rowp[4 + 2 * half];
        float t5 = rowp[5];
        a1.y = half ? 0.0f : t5;        // K=7 padded with zero via v_cndmask

        v8f d = __builtin_amdgcn_wmma_f32_16x16x4_f32(
            false, a0, false, b0, (short)0, cinit, false, false);
        d = __builtin_amdgcn_wmma_f32_16x16x4_f32(
            false, a1, false, b1, (short)0, d, false, false);

        // Column N=0 lives in lane 0 (M=0..7) and lane 16 (M=8..15).
        if (row16 == 0) {
            float4 o0, o1;
            o0.x = d[0] > THRESH ? PENALTY : 0.0f;
            o0.y = d[1] > THRESH ? PENALTY : 0.0f;
            o0.z = d[2] > THRESH ? PENALTY : 0.0f;
            o0.w = d[3] > THRESH ? PENALTY : 0.0f;
            o1.x = d[4] > THRESH ? PENALTY : 0.0f;
            o1.y = d[5] > THRESH ? PENALTY : 0.0f;
            o1.z = d[6] > THRESH ? PENALTY : 0.0f;
            o1.w = d[7] > THRESH ? PENALTY : 0.0f;
            float4* op = (float4*)(out + (size_t)tile * 16 + half * 8);
            op[0] = o0;
            op[1] = o1;
        }
    }
}

// Scalar tail for row counts not divisible by 16 (not hit at the given shapes).
__global__ void cost_tail_kernel(const float* __restrict__ sb,
                                 const float* __restrict__ ws,
                                 float* __restrict__ out,
                                 int row_start, int nrows) {
    int r = row_start + (int)(blockIdx.x * blockDim.x + threadIdx.x);
    if (r >= row_start + nrows) return;
    float fi = ws[7];
    const float* rowp = sb + (size_t)r * 7;
#pragma unroll
    for (int k = 0; k < 7; ++k) fi += ws[k] * rowp[k];
    out[r] = fi > THRESH ? PENALTY : 0.0f;
}

extern "C" void kernel_launch(void* const* d_in, const int* in_sizes, int n_in,
                              void* d_out, int out_size, void* d_ws, size_t ws_size,
                              hipStream_t stream) {
    const float* start_state = (const float*)d_in[0];  // [1,7]
    const float* state_batch = (const float*)d_in[1];  // [N,H,7]
    const float* normals     = (const float*)d_in[2];  // [C,3]
    const float* jac         = (const float*)d_in[3];  // [C,3,7]
    const float* forces      = (const float*)d_in[4];  // [C]
    float* out = (float*)d_out;
    float* ws  = (float*)d_ws;

    const int C     = in_sizes[4];
    const int rows  = in_sizes[1] / 7;    // N*H
    const int ntile = rows / 16;
    const int rem   = rows - ntile * 16;

    prep_kernel<<<1, 64, 0, stream>>>(start_state, normals, jac, forces, ws, C);

    if (ntile > 0) {
        int waves_needed = ntile;
        int blocks = (waves_needed * 32 + 255) / 256;
        if (blocks > 1024) blocks = 1024;
        cost_wmma_kernel<<<blocks, 256, 0, stream>>>(state_batch, ws, out, ntile);
    }
    if (rem > 0) {
        cost_tail_kernel<<<1, 32, 0, stream>>>(state_batch, ws, out,
                                               ntile * 16, rem);
    }
}